// SearchableQuantumModel_70050916598215
// MI455X (gfx1250) — compile-verified
//
#include <hip/hip_runtime.h>

#define NQ   12
#define DIM  4096          // 2^NQ
#define TPB  256
#define PER  (DIM / TPB)   // 16 amplitudes per thread

typedef float v8f __attribute__((ext_vector_type(8)));
typedef float v2f __attribute__((ext_vector_type(2)));
typedef __attribute__((address_space(3))) float lds_f;

// ---------------------------------------------------------------------------
// Kernel 1: build the six 16x16 group unitaries U = G(q0)(x)G(q1)(x)G(q2)(x)G(q3)
// for (layer l in 0..1) x (group g in 0..2).  Batch-independent, 1.5K floats.
// ---------------------------------------------------------------------------
__global__ void build_unitaries(const float* __restrict__ weights,
                                float* __restrict__ U /* [6][16][16] */) {
    const int mat = blockIdx.x;          // 0..5 = l*3 + g
    const int l = mat / 3, g = mat % 3;
    const int e = threadIdx.x;           // 0..255
    const int a = e >> 4, k = e & 15;
    float prod = 1.0f;
#pragma unroll
    for (int j = 0; j < 4; ++j) {
        const int q = g * 4 + j;                 // qubit index
        const float th = 0.5f * weights[l * NQ + q];
        const float c = __cosf(th), s = __sinf(th);
        const int ab = (a >> (3 - j)) & 1;       // qubit g*4+0 is MSB of group idx
        const int kb = (k >> (3 - j)) & 1;
        // RY: [[c,-s],[s,c]]
        const float gv = ab ? (kb ? c : s) : (kb ? -s : c);
        prod *= gv;
    }
    U[mat * 256 + e] = prod;
}

// ---------------------------------------------------------------------------
// Group unitary application (LDS -> LDS) via V_WMMA_F32_16X16X4_F32.
// Compile-time strides => base + immediate-offset DS ops.  Each wave owns 2
// independent 16x16 output tiles: A preloaded once, all B chunks fetched
// before the WMMA burst (two independent accumulator chains).
// ---------------------------------------------------------------------------
template <int SA, int SHI, int SLO>
__device__ __forceinline__ void apply_group(const float* __restrict__ Ug,
                                            const float* __restrict__ cur,
                                            float* __restrict__ nxt,
                                            int lane, int wave) {
    const int n    = lane & 15;   // B column within tile / A row
    const int half = lane >> 4;   // K/M half split per ISA VGPR layout

    v2f A[4];
#pragma unroll
    for (int kc = 0; kc < 4; ++kc) {
        const int kk = kc * 4 + half * 2;
        A[kc].x = Ug[n * 16 + kk];
        A[kc].y = Ug[n * 16 + kk + 1];
    }

    const int t0 = wave * 2, t1 = t0 + 1;
    const int b0 = t0 * SHI + n * SLO;
    const int b1 = t1 * SHI + n * SLO;

    v2f B0[4], B1[4];
#pragma unroll
    for (int kc = 0; kc < 4; ++kc) {
        const int kk = kc * 4 + half * 2;
        B0[kc].x = cur[kk * SA + b0];
        B0[kc].y = cur[(kk + 1) * SA + b0];
        B1[kc].x = cur[kk * SA + b1];
        B1[kc].y = cur[(kk + 1) * SA + b1];
    }

    v8f acc0 = {0.f, 0.f, 0.f, 0.f, 0.f, 0.f, 0.f, 0.f};
    v8f acc1 = {0.f, 0.f, 0.f, 0.f, 0.f, 0.f, 0.f, 0.f};
#pragma unroll
    for (int kc = 0; kc < 4; ++kc) {
        acc0 = __builtin_amdgcn_wmma_f32_16x16x4_f32(
            false, A[kc], false, B0[kc], (short)0, acc0, false, false);
        acc1 = __builtin_amdgcn_wmma_f32_16x16x4_f32(
            false, A[kc], false, B1[kc], (short)0, acc1, false, false);
    }

#pragma unroll
    for (int v = 0; v < 8; ++v) {
        nxt[(v + 8 * half) * SA + b0] = acc0[v];
        nxt[(v + 8 * half) * SA + b1] = acc1[v];
    }
}

// Group 0 with the CNOT-ring permutation folded into the B gather: B elements
// are read from per-lane precomputed permuted indices (psrc), saving a full
// statevector permutation round-trip + barrier per layer.
__device__ __forceinline__ void apply_group0_perm(const float* __restrict__ Ug,
                                                  const float* __restrict__ cur,
                                                  float* __restrict__ nxt,
                                                  int lane, int wave,
                                                  const int* __restrict__ psrc) {
    const int n    = lane & 15;
    const int half = lane >> 4;

    v2f A[4];
#pragma unroll
    for (int kc = 0; kc < 4; ++kc) {
        const int kk = kc * 4 + half * 2;
        A[kc].x = Ug[n * 16 + kk];
        A[kc].y = Ug[n * 16 + kk + 1];
    }

    v2f B0[4], B1[4];
#pragma unroll
    for (int kc = 0; kc < 4; ++kc) {
        B0[kc].x = cur[psrc[kc * 2 + 0]];
        B0[kc].y = cur[psrc[kc * 2 + 1]];
        B1[kc].x = cur[psrc[8 + kc * 2 + 0]];
        B1[kc].y = cur[psrc[8 + kc * 2 + 1]];
    }

    v8f acc0 = {0.f, 0.f, 0.f, 0.f, 0.f, 0.f, 0.f, 0.f};
    v8f acc1 = {0.f, 0.f, 0.f, 0.f, 0.f, 0.f, 0.f, 0.f};
#pragma unroll
    for (int kc = 0; kc < 4; ++kc) {
        acc0 = __builtin_amdgcn_wmma_f32_16x16x4_f32(
            false, A[kc], false, B0[kc], (short)0, acc0, false, false);
        acc1 = __builtin_amdgcn_wmma_f32_16x16x4_f32(
            false, A[kc], false, B1[kc], (short)0, acc1, false, false);
    }

    // D layout for group 0: axis stride 256, tile stride 16, col stride 1
    const int t0 = wave * 2, t1 = t0 + 1;
    const int b0 = t0 * 16 + n;
    const int b1 = t1 * 16 + n;
#pragma unroll
    for (int v = 0; v < 8; ++v) {
        nxt[(v + 8 * half) * 256 + b0] = acc0[v];
        nxt[(v + 8 * half) * 256 + b1] = acc1[v];
    }
}

// ---------------------------------------------------------------------------
// Kernel 2: one workgroup (8 wave32) per batch element.  16KB statevector
// double-buffered in LDS; per layer: (perm-fused) group0, group1, group2
// 16x16 (x) 16x256 GEMMs on the matrix pipe.  Unitary table staged by the
// async global->LDS engine, overlapped with initial-state VALU work.
// ---------------------------------------------------------------------------
__global__ void __launch_bounds__(TPB)
qsim_kernel(const float* __restrict__ x,     // [B, 12]
            const float* __restrict__ U6,    // [6, 16, 16]
            const float* __restrict__ W,     // [2, 12]
            const float* __restrict__ bias,  // [2]
            float* __restrict__ out)         // [B, 2]
{
    __shared__ float bufA[DIM];
    __shared__ float bufB[DIM];
    __shared__ float Ush[6 * 256];
    __shared__ float red[16];

    const int tid  = threadIdx.x;
    const int bid  = blockIdx.x;
    const int lane = tid & 31;
    const int wave = tid >> 5;

    // async DMA the unitary table straight into LDS (no VGPR round-trip),
    // tracked on ASYNCcnt; overlaps with the trig + product-state VALU below.
#pragma unroll
    for (int k = 0; k < 6; ++k) {
        const unsigned goff = 4u * (unsigned)(tid + k * TPB);
        asm volatile("global_load_async_to_lds_b32 %0, %1, %2"
                     :: "v"((lds_f*)&Ush[tid + k * TPB]), "v"(goff), "s"(U6)
                     : "memory");
    }

    // per-qubit cos/sin of x/2
    float cq[NQ], sq[NQ];
#pragma unroll
    for (int q = 0; q < NQ; ++q) {
        const float th = 0.5f * x[bid * NQ + q];
        cq[q] = __cosf(th);
        sq[q] = __sinf(th);
    }

    // initial product state: amp[idx] = prod_q (bit_q ? sin : cos), qubit 0 = MSB
#pragma unroll
    for (int j = 0; j < PER; ++j) {
        const int idx = tid + j * TPB;
        float amp = 1.0f;
#pragma unroll
        for (int q = 0; q < NQ; ++q)
            amp *= ((idx >> (NQ - 1 - q)) & 1) ? sq[q] : cq[q];
        bufA[idx] = amp;
    }

    // Precompute the permuted source index for each B element this lane feeds
    // into the group-0 WMMAs.  CNOT-ring composite: src = c1(c2(...c12(i))),
    // each map flips the target bit iff the control bit is set.
    int psrc[16];
#pragma unroll
    for (int tt = 0; tt < 2; ++tt) {
#pragma unroll
        for (int kc = 0; kc < 4; ++kc) {
#pragma unroll
            for (int j = 0; j < 2; ++j) {
                const int kk = kc * 4 + (lane >> 4) * 2 + j;
                int i = kk * 256 + (wave * 2 + tt) * 16 + (lane & 15);
#pragma unroll
                for (int p = NQ - 1; p >= 0; --p) {
                    const int ci = p, ti = (p + 1) % NQ;
                    const int pc = NQ - 1 - ci, pt = NQ - 1 - ti;
                    i ^= ((i >> pc) & 1) << pt;
                }
                psrc[tt * 8 + kc * 2 + j] = i;
            }
        }
    }

    // drain the async DMA before any wave consumes Ush
    asm volatile("s_wait_asynccnt 0" ::: "memory");
    __syncthreads();

    float* cur = bufA;
    float* nxt = bufB;

#pragma unroll
    for (int l = 0; l < 2; ++l) {
        // group 0 (qubits 0-3) with CNOT ring fused into the B gather
        apply_group0_perm(&Ush[(l * 3 + 0) * 256], cur, nxt, lane, wave, psrc);
        { float* t = cur; cur = nxt; nxt = t; }
        __syncthreads();

        // group 1: qubits 4-7 (axis stride 16)
        apply_group<16, 256, 1>(&Ush[(l * 3 + 1) * 256], cur, nxt, lane, wave);
        { float* t = cur; cur = nxt; nxt = t; }
        __syncthreads();

        // group 2: qubits 8-11 (axis stride 1)
        apply_group<1, 256, 16>(&Ush[(l * 3 + 2) * 256], cur, nxt, lane, wave);
        { float* t = cur; cur = nxt; nxt = t; }
        __syncthreads();
    }

    // ---- readout: out[b,o] = sum_i cur[i]^2 * M[i][o] + bias[o],
    //      M[i][o] = sum_q (1-2*bit_q(i)) * W[o][q]  (folded signs@W).
    // idx = tid + j*256: tid drives qubits 4-11, j drives qubits 0-3, so
    // split M into per-thread base (hoisted) + per-j delta (constant signs).
    float w0[NQ], w1[NQ];
#pragma unroll
    for (int q = 0; q < NQ; ++q) { w0[q] = W[q]; w1[q] = W[NQ + q]; }

    float base0 = 0.f, base1 = 0.f;
#pragma unroll
    for (int q = 4; q < NQ; ++q) {
        const float sgn = ((tid >> (NQ - 1 - q)) & 1) ? -1.f : 1.f;
        base0 += sgn * w0[q];
        base1 += sgn * w1[q];
    }

    float a0 = 0.f, a1 = 0.f;
#pragma unroll
    for (int j = 0; j < PER; ++j) {
        const float v = cur[tid + j * TPB];
        const float p = v * v;
        float m0 = base0, m1 = base1;
#pragma unroll
        for (int q = 0; q < 4; ++q) {
            const float sgn = ((j >> (3 - q)) & 1) ? -1.f : 1.f;  // compile-time
            m0 += sgn * w0[q];
            m1 += sgn * w1[q];
        }
        a0 += p * m0;
        a1 += p * m1;
    }

    // wave-level reduction on the lane-permute network, then one tiny pass
#pragma unroll
    for (int m = 16; m >= 1; m >>= 1) {
        a0 += __shfl_xor(a0, m, 32);
        a1 += __shfl_xor(a1, m, 32);
    }
    if (lane == 0) { red[wave] = a0; red[8 + wave] = a1; }
    __syncthreads();
    if (tid == 0) {
        float r0 = bias[0], r1 = bias[1];
#pragma unroll
        for (int w = 0; w < 8; ++w) { r0 += red[w]; r1 += red[8 + w]; }
        out[bid * 2 + 0] = r0;
        out[bid * 2 + 1] = r1;
    }
}

// ---------------------------------------------------------------------------
extern "C" void kernel_launch(void* const* d_in, const int* in_sizes, int n_in,
                              void* d_out, int out_size, void* d_ws, size_t ws_size,
                              hipStream_t stream) {
    const float* x       = (const float*)d_in[0];   // [B, 12]
    const float* weights = (const float*)d_in[1];   // [2, 12]
    const float* W       = (const float*)d_in[2];   // [2, 12]
    const float* b       = (const float*)d_in[3];   // [2]
    float* out = (float*)d_out;

    const int B = in_sizes[0] / NQ;
    float* U6 = (float*)d_ws;                       // 6*256 floats of scratch

    build_unitaries<<<6, TPB, 0, stream>>>(weights, U6);
    qsim_kernel<<<B, TPB, 0, stream>>>(x, U6, W, b, out);
}